// ChamferDistance_27805618274644
// MI455X (gfx1250) — compile-verified
//
#include <hip/hip_runtime.h>
#include <math.h>

typedef float v2f __attribute__((ext_vector_type(2)));
typedef float v8f __attribute__((ext_vector_type(8)));

#define BIG 1e10f

// One direction of Chamfer: for each query point q of batch b,
//   out[b*NQ + q] = (q < lenQ[b]) ? sqrt(max(min_{r<lenR[b]} d2(q,r), 0)) : 0
//
// Formulation: d2 = |q|^2 + |r|^2 - 2 q.r.  The min over r of
// (|r|^2 - 2 q.r) is independent of |q|^2, and adding |q|^2 >= 0 then
// clamping at 0 equals clamping the full d2 at 0 only up to the |q|^2
// shift -- so |q|^2 stays out of the product:
//   A row (M=query)   = [qx, qy, 1, 0]              (K = 4, f32)
//   B col (N=ref)     = [-2*rx, -2*ry, rr+qq? no -> rr|BIG, 0]
//   D[q][r] = rr - 2 q.r, then add qq at the end before clamp/sqrt.
// Invalid refs get column [-2rx,-2ry,BIG,0] => D = BIG - 2 q.r (~BIG).
//
// V_WMMA_F32_16X16X4_F32 layouts (wave32):
//   A 16x4 : lanes 0-15 hold row M=lane, vgpr0=K0, vgpr1=K1;
//            lanes 16-31 hold K2,K3 of row M=lane-16.
//   B 4x16 : mirrored: lanes 0-15 vgpr0=K0 row, vgpr1=K1 row;
//            lanes 16-31 vgpr0=K2 row, vgpr1=K3 row.  (N = lane&15)
//   D 16x16: vgpr i, lanes 0-15 -> (M=i, N=lane); lanes 16-31 -> (M=i+8).
__global__ __launch_bounds__(256) void chamfer_dir_kernel(
    const float* __restrict__ query,  // [B, NQ, 2]
    const float* __restrict__ ref,    // [B, NR, 2]
    const int*  __restrict__ lenQ,    // [B]
    const int*  __restrict__ lenR,    // [B]
    float* __restrict__ out,          // [B, NQ]
    int NQ, int NR)
{
    extern __shared__ char smem_raw[];
    float4* sRef = (float4*)smem_raw;  // NR entries: {-2x, -2y, rr|BIG, 0}

    const int groups = NQ >> 7;              // strip-groups per batch (8 strips each)
    const int b = blockIdx.x / groups;
    const int g = blockIdx.x % groups;

    // ---- stage folded reference columns into LDS (once per block) ----
    const float2* Rb = (const float2*)ref + (size_t)b * NR;
    const int lr = lenR[b];
    for (int j = threadIdx.x; j < NR; j += blockDim.x) {
        float2 p = Rb[j];
        float rr = fmaf(p.x, p.x, p.y * p.y);
        float4 v;
        v.x = -2.0f * p.x;
        v.y = -2.0f * p.y;
        v.z = (j < lr) ? rr : BIG;
        v.w = 0.0f;
        sRef[j] = v;
    }
    __syncthreads();

    // ---- per-wave: one strip of 16 query points ----
    const int lane = threadIdx.x & 31;
    const int wid  = threadIdx.x >> 5;
    const int half = lane >> 4;   // 0: K0/K1 lanes, 1: K2/K3 lanes
    const int l15  = lane & 15;
    const int q0   = (g * 8 + wid) * 16;

    const float2* Qb = (const float2*)query + (size_t)b * NQ;
    float2 qp = Qb[q0 + l15];                 // always in-bounds (mask is logical)
    v2f a;
    a.x = half ? 1.0f : qp.x;                 // K0 | K2
    a.y = half ? 0.0f : qp.y;                 // K1 | K3

    v8f acc;
#pragma unroll
    for (int i = 0; i < 8; ++i) acc[i] = BIG;

    const float* sR = (const float*)sRef;
    for (int t = 0; t < NR; t += 16) {
        // lanes 0-15 read {-2x,-2y} of ref t+l15; lanes 16-31 read {rr|BIG, 0}
        const float2 bb = *(const float2*)(sR + (size_t)(t + l15) * 4 + half * 2);
        v2f bv; bv.x = bb.x; bv.y = bb.y;
        v8f cz = {};
        v8f d = __builtin_amdgcn_wmma_f32_16x16x4_f32(
            false, a, false, bv, (short)0, cz, false, false);
#pragma unroll
        for (int i = 0; i < 8; ++i) acc[i] = fminf(acc[i], d[i]);
    }

    // ---- cross-lane min over the 16 ref lanes of each half ----
#pragma unroll
    for (int i = 0; i < 8; ++i) {
        float v = acc[i];
        v = fminf(v, __shfl_xor(v, 1, 32));
        v = fminf(v, __shfl_xor(v, 2, 32));
        v = fminf(v, __shfl_xor(v, 4, 32));
        v = fminf(v, __shfl_xor(v, 8, 32));
        acc[i] = v;
    }

    const int lq = lenQ[b];
    if (l15 == 0) {  // lanes 0 and 16 each hold 8 query results
#pragma unroll
        for (int i = 0; i < 8; ++i) {
            int q = q0 + half * 8 + i;
            float2 p = Qb[q];
            float qq = fmaf(p.x, p.x, p.y * p.y);      // add back |q|^2
            float v = fmaxf(acc[i] + qq, 0.0f);
            v = sqrtf(v);
            out[(size_t)b * NQ + q] = (q < lq) ? v : 0.0f;
        }
    }
}

extern "C" void kernel_launch(void* const* d_in, const int* in_sizes, int n_in,
                              void* d_out, int out_size, void* d_ws, size_t ws_size,
                              hipStream_t stream) {
    const float* src  = (const float*)d_in[0];   // [B, N, 2]
    const float* tgt  = (const float*)d_in[1];   // [B, M, 2]
    const int*   slen = (const int*)d_in[2];     // [B]
    const int*   tlen = (const int*)d_in[3];     // [B]
    float* out = (float*)d_out;                  // [B*N] fwd ++ [B*M] bwd

    const int B = in_sizes[2];
    const int N = in_sizes[0] / (B * 2);
    const int M = in_sizes[1] / (B * 2);

    // forward: queries = source, refs = target
    {
        dim3 grid(B * (N >> 7));
        size_t lds = (size_t)M * sizeof(float4);
        chamfer_dir_kernel<<<grid, 256, lds, stream>>>(
            src, tgt, slen, tlen, out, N, M);
    }
    // backward: queries = target, refs = source
    {
        dim3 grid(B * (M >> 7));
        size_t lds = (size_t)N * sizeof(float4);
        chamfer_dir_kernel<<<grid, 256, lds, stream>>>(
            tgt, src, tlen, slen, out + (size_t)B * N, M, N);
    }
}